// NGCF_69123203662125
// MI455X (gfx1250) — compile-verified
//
#include <hip/hip_runtime.h>
#include <hip/hip_bf16.h>

#define U_CNT 100000
#define I_CNT 50000
#define D_DIM 64
#define L_CNT 3
#define E_CNT 2000000
#define N_CNT (U_CNT + I_CNT)   // 150000
#define NTILES (N_CNT / 16)     // 9375 exactly

typedef __attribute__((ext_vector_type(2))) float v2f;
typedef __attribute__((ext_vector_type(8))) float v8f;

// ---------------------------------------------------------------------------
// 1. init: X_cur = concat(u_emb, i_emb); acc(d_out) = same
// ---------------------------------------------------------------------------
__global__ __launch_bounds__(256) void ngcf_init(const float* __restrict__ u_emb,
                                                 const float* __restrict__ i_emb,
                                                 float* __restrict__ X,
                                                 float* __restrict__ acc) {
    size_t idx = (size_t)blockIdx.x * blockDim.x + threadIdx.x;
    if (idx >= (size_t)N_CNT * D_DIM) return;
    float v = (idx < (size_t)U_CNT * D_DIM) ? u_emb[idx] : i_emb[idx - (size_t)U_CNT * D_DIM];
    X[idx] = v;
    acc[idx] = v;
}

// ---------------------------------------------------------------------------
// 2. degrees (self-loop baked in as 1.0) -> d^{-1/2}
// ---------------------------------------------------------------------------
__global__ __launch_bounds__(256) void deg_init(float* __restrict__ deg) {
    int n = blockIdx.x * blockDim.x + threadIdx.x;
    if (n < N_CNT) deg[n] = 1.0f;
}

__global__ __launch_bounds__(256) void deg_edges(const int* __restrict__ edge_u,
                                                 const int* __restrict__ edge_i,
                                                 float* __restrict__ deg) {
    int e = blockIdx.x * blockDim.x + threadIdx.x;
    if (e >= E_CNT) return;
    __hip_atomic_fetch_add(&deg[edge_u[e]], 1.0f, __ATOMIC_RELAXED, __HIP_MEMORY_SCOPE_AGENT);
    __hip_atomic_fetch_add(&deg[U_CNT + edge_i[e]], 1.0f, __ATOMIC_RELAXED, __HIP_MEMORY_SCOPE_AGENT);
}

__global__ __launch_bounds__(256) void dinv_kernel(const float* __restrict__ deg,
                                                   float* __restrict__ dinv) {
    int n = blockIdx.x * blockDim.x + threadIdx.x;
    if (n < N_CNT) dinv[n] = rsqrtf(fmaxf(deg[n], 1.0f));
}

// ---------------------------------------------------------------------------
// 3a. SpMM self-loop term: G[n,:] = dinv[n]^2 * X[n,:]
// ---------------------------------------------------------------------------
__global__ __launch_bounds__(256) void spmm_self(const float* __restrict__ X,
                                                 const float* __restrict__ dinv,
                                                 float* __restrict__ G) {
    size_t idx = (size_t)blockIdx.x * blockDim.x + threadIdx.x;
    if (idx >= (size_t)N_CNT * D_DIM) return;
    float d = dinv[idx >> 6];
    G[idx] = d * d * X[idx];
}

// ---------------------------------------------------------------------------
// 3b. SpMM edges: one wave32 per edge; lanes cover dims {l, l+32}; both dirs.
// ---------------------------------------------------------------------------
__global__ __launch_bounds__(256) void spmm_edges(const float* __restrict__ X,
                                                  const float* __restrict__ dinv,
                                                  const int* __restrict__ edge_u,
                                                  const int* __restrict__ edge_i,
                                                  float* __restrict__ G) {
    size_t gtid = (size_t)blockIdx.x * blockDim.x + threadIdx.x;
    size_t e = gtid >> 5;
    int lane = (int)(gtid & 31);
    if (e >= E_CNT) return;
    int u = edge_u[e];
    int v = U_CNT + edge_i[e];
    float c = dinv[u] * dinv[v];
    const float* xu = X + (size_t)u * D_DIM;
    const float* xv = X + (size_t)v * D_DIM;
    float* gu = G + (size_t)u * D_DIM;
    float* gv = G + (size_t)v * D_DIM;
    __hip_atomic_fetch_add(&gv[lane],      c * xu[lane],      __ATOMIC_RELAXED, __HIP_MEMORY_SCOPE_AGENT);
    __hip_atomic_fetch_add(&gv[lane + 32], c * xu[lane + 32], __ATOMIC_RELAXED, __HIP_MEMORY_SCOPE_AGENT);
    __hip_atomic_fetch_add(&gu[lane],      c * xv[lane],      __ATOMIC_RELAXED, __HIP_MEMORY_SCOPE_AGENT);
    __hip_atomic_fetch_add(&gu[lane + 32], c * xv[lane + 32], __ATOMIC_RELAXED, __HIP_MEMORY_SCOPE_AGENT);
}

// ---------------------------------------------------------------------------
// 4. Dense layer: per wave, one 16-row tile.
//    sum_e = LReLU(G Wgc^T + bgc); bi_e = LReLU((X*G) Wbi^T + bbi)
//    out   = L2normalize(sum_e + bi_e);  Xn = out; acc += out
//    Uses V_WMMA_F32_16X16X4_F32 (f32 A 16x4 / B 4x16 / C,D 16x16).
// ---------------------------------------------------------------------------
__global__ __launch_bounds__(256) void dense_layer(const float* __restrict__ G,
                                                   const float* __restrict__ X,
                                                   const float* __restrict__ Wgc,
                                                   const float* __restrict__ bgc,
                                                   const float* __restrict__ Wbi,
                                                   const float* __restrict__ bbi,
                                                   float* __restrict__ Xn,
                                                   float* __restrict__ acc) {
    __shared__ float lWgc[64 * 64];
    __shared__ float lWbi[64 * 64];

    // stage weights (row-major [j][k]) into LDS, coalesced
    for (int i = threadIdx.x; i < 64 * 64; i += 256) {
        lWgc[i] = Wgc[i];
        lWbi[i] = Wbi[i];
    }
    __syncthreads();

    int waveInBlock = threadIdx.x >> 5;
    int lane = threadIdx.x & 31;
    int row16 = lane & 15;   // A row / B column selector within half-wave
    int hi = lane >> 4;      // 0: K pair {k0,k0+1}; 1: K pair {k0+2,k0+3}
    int tile = blockIdx.x * 8 + waveInBlock;
    if (tile >= NTILES) return;

    size_t rowBase = (size_t)tile * 16;
    const float* gRow = G + (rowBase + row16) * D_DIM;
    const float* xRow = X + (rowBase + row16) * D_DIM;

    v8f accGc[4], accBi[4];
#pragma unroll
    for (int t = 0; t < 4; ++t) { accGc[t] = (v8f)0.0f; accBi[t] = (v8f)0.0f; }

#pragma unroll
    for (int k0 = 0; k0 < 64; k0 += 4) {
        int kA = k0 + 2 * hi;
        v2f aG = *(const v2f*)(gRow + kA);  // A frag of G   (rows M=0..15)
        v2f aX = *(const v2f*)(xRow + kA);
        v2f aB = aG * aX;                   // A frag of X.*G
#pragma unroll
        for (int t = 0; t < 4; ++t) {
            int j = t * 16 + row16;         // output column
            v2f bGc = *(const v2f*)(&lWgc[j * 64 + kA]);  // B[k][j] = W[j][k]
            v2f bBi = *(const v2f*)(&lWbi[j * 64 + kA]);
            accGc[t] = __builtin_amdgcn_wmma_f32_16x16x4_f32(
                false, aG, false, bGc, (short)0, accGc[t], false, false);
            accBi[t] = __builtin_amdgcn_wmma_f32_16x16x4_f32(
                false, aB, false, bBi, (short)0, accBi[t], false, false);
        }
    }

    // biases per output column
    float bGcr[4], bBir[4];
#pragma unroll
    for (int t = 0; t < 4; ++t) {
        bGcr[t] = bgc[t * 16 + row16];
        bBir[t] = bbi[t * 16 + row16];
    }

    // bias + leaky relu(0.2) + sum; accumulate row sum-of-squares
    v8f outv[4];
    float ss[8];
#pragma unroll
    for (int r = 0; r < 8; ++r) ss[r] = 0.0f;
#pragma unroll
    for (int t = 0; t < 4; ++t) {
#pragma unroll
        for (int r = 0; r < 8; ++r) {
            float a = accGc[t][r] + bGcr[t];
            a = (a > 0.0f) ? a : 0.2f * a;
            float b = accBi[t][r] + bBir[t];
            b = (b > 0.0f) ? b : 0.2f * b;
            float o = a + b;
            outv[t][r] = o;
            ss[r] += o * o;
        }
    }

    // reduce across the 16 lanes of each half-wave group (row m = r + 8*hi)
    float inv[8];
#pragma unroll
    for (int r = 0; r < 8; ++r) {
        float s = ss[r];
        s += __shfl_xor(s, 1, 32);
        s += __shfl_xor(s, 2, 32);
        s += __shfl_xor(s, 4, 32);
        s += __shfl_xor(s, 8, 32);
        inv[r] = 1.0f / fmaxf(sqrtf(s), 1e-12f);
    }

    // normalize, store next-layer X and accumulate into acc
#pragma unroll
    for (int t = 0; t < 4; ++t) {
#pragma unroll
        for (int r = 0; r < 8; ++r) {
            size_t node = rowBase + (size_t)(r + 8 * hi);
            size_t off = node * D_DIM + (size_t)(t * 16 + row16);
            float o = outv[t][r] * inv[r];
            Xn[off] = o;
            acc[off] += o;
        }
    }
}

// ---------------------------------------------------------------------------
// 5. final mean over (L+1) layer outputs
// ---------------------------------------------------------------------------
__global__ __launch_bounds__(256) void finalize(float* __restrict__ acc) {
    size_t idx = (size_t)blockIdx.x * blockDim.x + threadIdx.x;
    if (idx < (size_t)N_CNT * D_DIM) acc[idx] *= 1.0f / (float)(L_CNT + 1);
}

// ---------------------------------------------------------------------------
extern "C" void kernel_launch(void* const* d_in, const int* in_sizes, int n_in,
                              void* d_out, int out_size, void* d_ws, size_t ws_size,
                              hipStream_t stream) {
    const float* u_emb = (const float*)d_in[0];
    const float* i_emb = (const float*)d_in[1];
    const float* W_gc  = (const float*)d_in[2];
    const float* b_gc  = (const float*)d_in[3];
    const float* W_bi  = (const float*)d_in[4];
    const float* b_bi  = (const float*)d_in[5];
    const int*   edge_u = (const int*)d_in[6];
    const int*   edge_i = (const int*)d_in[7];
    float* acc = (float*)d_out;   // running sum of layer outputs, scaled at end

    // workspace carve-up (floats)
    float* ws   = (float*)d_ws;
    float* deg  = ws;                               // N
    float* dinv = deg + N_CNT;                      // N
    float* Xa   = dinv + N_CNT;                     // N*64
    float* Xb   = Xa + (size_t)N_CNT * D_DIM;       // N*64
    float* Gbuf = Xb + (size_t)N_CNT * D_DIM;       // N*64

    const size_t ND = (size_t)N_CNT * D_DIM;
    const int nd_blocks = (int)((ND + 255) / 256);
    const int n_blocks  = (N_CNT + 255) / 256;
    const int e_blocks  = (E_CNT + 255) / 256;
    const int ew_blocks = (int)(((size_t)E_CNT * 32 + 255) / 256);
    const int dn_blocks = (NTILES + 7) / 8;

    ngcf_init<<<nd_blocks, 256, 0, stream>>>(u_emb, i_emb, Xa, acc);
    deg_init<<<n_blocks, 256, 0, stream>>>(deg);
    deg_edges<<<e_blocks, 256, 0, stream>>>(edge_u, edge_i, deg);
    dinv_kernel<<<n_blocks, 256, 0, stream>>>(deg, dinv);

    float* Xc = Xa;
    float* Xn = Xb;
    for (int l = 0; l < L_CNT; ++l) {
        spmm_self<<<nd_blocks, 256, 0, stream>>>(Xc, dinv, Gbuf);
        spmm_edges<<<ew_blocks, 256, 0, stream>>>(Xc, dinv, edge_u, edge_i, Gbuf);
        dense_layer<<<dn_blocks, 256, 0, stream>>>(
            Gbuf, Xc,
            W_gc + (size_t)l * 64 * 64, b_gc + (size_t)l * 64,
            W_bi + (size_t)l * 64 * 64, b_bi + (size_t)l * 64,
            Xn, acc);
        float* tmp = Xc; Xc = Xn; Xn = tmp;
    }
    finalize<<<nd_blocks, 256, 0, stream>>>(acc);
}